// SplineInterpolateBase_39822936769134
// MI455X (gfx1250) — compile-verified
//
#include <hip/hip_runtime.h>

typedef __attribute__((ext_vector_type(2))) float v2f;
typedef __attribute__((ext_vector_type(4))) float v4f;
typedef __attribute__((ext_vector_type(8))) float v8f;

#define N_X   4096
#define KDEG  3
#define N_T   4100                      // knots = N + 2*(k+1) - ... = 4092 interior + 8
#define M_B   4096                      // basis columns = N_T - k - 1
#define BASIS_OFF 4100
#define BTB_OFF   (4100 + (size_t)4096 * 4096)

// ---------------- Kernel 1: FITPACK-style knots ----------------
__global__ void knots_kernel(const float* __restrict__ x, float* __restrict__ t) {
    int i = blockIdx.x * blockDim.x + threadIdx.x;
    if (i >= N_T) return;
    float v;
    if (i < KDEG + 1)        v = x[0];
    else if (i >= N_X)       v = x[N_X - 1];
    else                     v = ((x[i - 3] + x[i - 2]) + x[i - 1]) * (1.0f / 3.0f);
    t[i] = v;
}

// ---------------- Kernel 2: zero-fill basis + BTB (bandwidth-bound) ----------------
__global__ void fill_zero_kernel(float* __restrict__ p, long n4) {
    long i = (long)blockIdx.x * blockDim.x + threadIdx.x;
    if (i < n4) ((v4f*)p)[i] = (v4f){0.f, 0.f, 0.f, 0.f};
}

// ---------------- Kernel 3: per-row de Boor band values + start-column ----------------
__global__ void basis_kernel(const float* __restrict__ x,
                             const float* __restrict__ t,
                             float* __restrict__ basis,
                             int* __restrict__ cstart) {
    int r = blockIdx.x * blockDim.x + threadIdx.x;
    if (r >= N_X) return;
    float xv = x[r];

    // interval: largest j in [k, m-1] with t[j] <= xv (clamped at top edge)
    int lo = KDEG, hi = M_B - 1;
    while (lo < hi) {
        int mid = (lo + hi + 1) >> 1;
        if (t[mid] <= xv) lo = mid; else hi = mid - 1;
    }
    int j = lo;

    // Cox–de Boor with 0/0 -> 0 (matches _safe_div semantics)
    float Nv[KDEG + 1];
    Nv[0] = 1.0f;
    for (int d = 1; d <= KDEG; ++d) {
        float saved = 0.0f;
        for (int rr = 0; rr < d; ++rr) {
            float den  = t[j + rr + 1] - t[j + rr + 1 - d];
            float temp = (den != 0.0f) ? (Nv[rr] / den) : 0.0f;
            Nv[rr] = saved + (t[j + rr + 1] - xv) * temp;
            saved  = (xv - t[j + rr + 1 - d]) * temp;
        }
        Nv[d] = saved;
    }

    int c0 = j - KDEG;                      // in [0, 4092]
    cstart[r] = c0;
    size_t rowbase = (size_t)r * M_B;
    #pragma unroll
    for (int i = 0; i <= KDEG; ++i) basis[rowbase + c0 + i] = Nv[i];
}

// ---------------- Kernel 4: banded B^T B via V_WMMA_F32_16X16X4_F32 ----------------
// One wave per 16x16 output tile; only tile-diagonals |I-J| <= 1 are non-zero.
__global__ void __launch_bounds__(32)
btb_tile_kernel(const float* __restrict__ basis,
                const int*   __restrict__ cstart,
                float*       __restrict__ btb,
                const float* __restrict__ s_ptr) {
    int tileId = blockIdx.x;
    int I  = tileId / 3;
    int dJ = (tileId % 3) - 1;
    int J  = I + dJ;
    if (J < 0 || J >= M_B / 16) return;       // wave-uniform
    int lane = threadIdx.x;
    float s = *s_ptr;

    // rows touching column block B: cstart in [16B-3, 16B+15]; cstart is monotone.
    auto lower_bound = [&](int val) {         // first r with cstart[r] >= val
        int a = 0, b = N_X;
        while (a < b) { int m = (a + b) >> 1; if (cstart[m] < val) a = m + 1; else b = m; }
        return a;
    };
    auto upper_bound = [&](int val) {         // first r with cstart[r] > val
        int a = 0, b = N_X;
        while (a < b) { int m = (a + b) >> 1; if (cstart[m] <= val) a = m + 1; else b = m; }
        return a;
    };
    int loI = lower_bound(16 * I - 3), hiI = upper_bound(16 * I + 15) - 1;
    int loJ = lower_bound(16 * J - 3), hiJ = upper_bound(16 * J + 15) - 1;
    int rlo = max(loI, loJ), rhi = min(hiI, hiJ);

    v8f acc = (v8f){0.f, 0.f, 0.f, 0.f, 0.f, 0.f, 0.f, 0.f};

    int colA = 16 * I + (lane & 15);          // A: 16x4 f32, lanes 0-15 K=0,1; 16-31 K=2,3
    int colB = 16 * J + (lane & 15);
    int klo  = (lane >> 4) << 1;              // 0 or 2

    if (rlo <= rhi) {
        for (int kk = rlo & ~3; kk <= rhi; kk += 4) {   // kk <= 4092 always
            v2f a, b;
            a.x = basis[(size_t)(kk + klo)     * M_B + colA];
            a.y = basis[(size_t)(kk + klo + 1) * M_B + colA];
            b.x = basis[(size_t)(kk + klo)     * M_B + colB];
            b.y = basis[(size_t)(kk + klo + 1) * M_B + colB];
            acc = __builtin_amdgcn_wmma_f32_16x16x4_f32(
                false, a, false, b, (short)0, acc, false, false);
        }
    }

    // + s * I on diagonal tiles (C layout: M = v + 8*(lane>=16), N = lane&15)
    if (dJ == 0) {
        int n = lane & 15, mbase = (lane >> 4) * 8;
        #pragma unroll
        for (int v = 0; v < 8; ++v)
            if (n == mbase + v) acc[v] += s;
    }

    #pragma unroll
    for (int v = 0; v < 8; ++v) {
        size_t row = (size_t)(16 * I + v + 8 * (lane >> 4));
        btb[row * M_B + 16 * J + (lane & 15)] = acc[v];
    }
}

// ---------------- Launch ----------------
extern "C" void kernel_launch(void* const* d_in, const int* in_sizes, int n_in,
                              void* d_out, int out_size, void* d_ws, size_t ws_size,
                              hipStream_t stream) {
    const float* x     = (const float*)d_in[0];
    const float* s_ptr = (const float*)d_in[1];
    // d_in[2] is k == 3 (compile-time constant here)

    float* out   = (float*)d_out;
    float* knots = out;
    float* basis = out + BASIS_OFF;
    float* btb   = out + BTB_OFF;
    int*   cstart = (int*)d_ws;               // 4096 ints = 16 KB scratch

    knots_kernel<<<(N_T + 255) / 256, 256, 0, stream>>>(x, knots);

    long n4 = (long)2 * 4096 * 4096 / 4;      // both matrices, contiguous, 16B aligned
    fill_zero_kernel<<<(int)((n4 + 255) / 256), 256, 0, stream>>>(basis, n4);

    basis_kernel<<<N_X / 256, 256, 0, stream>>>(x, knots, basis, cstart);

    btb_tile_kernel<<<3 * (M_B / 16), 32, 0, stream>>>(basis, cstart, btb, s_ptr);
}